// TVHMM_50105088475512
// MI455X (gfx1250) — compile-verified
//
#include <hip/hip_runtime.h>
#include <math.h>

typedef __attribute__((ext_vector_type(2))) float v2f;
typedef __attribute__((ext_vector_type(8))) float v8f;

#define T_LEN 8192
#define KST   64
#define DX    128
#define DY    64
#define KK    4096   // K*K flattened transition columns

// ---------------------------------------------------------------------------
// Kernel 0: diagonal-Gaussian emission log-densities  log_em[T,K]
// ---------------------------------------------------------------------------
__global__ void __launch_bounds__(256)
log_em_kernel(const float* __restrict__ Y,
              const float* __restrict__ mu,
              const float* __restrict__ lvu,
              float* __restrict__ log_em) {
  int tid = threadIdx.x;
  int k = tid & 63;
  int t = blockIdx.x * 4 + (tid >> 6);
  const float LOG2PI = 1.8378770664093453f;
  float slv = 0.f, quad = 0.f;
  for (int d = 0; d < DY; ++d) {
    float lv = lvu[k * DY + d];
    lv = fminf(6.f, fmaxf(-6.f, lv));
    slv += lv;
    float diff = Y[t * DY + d] - mu[k * DY + d];
    quad += diff * diff * __expf(-lv);
  }
  log_em[t * KST + k] = -0.5f * ((float)DY * LOG2PI + slv + quad);
}

// ---------------------------------------------------------------------------
// Kernel 1: logits[t, i*64+j] = b[i,j] + X[t,:] . W[i,j,:]
// GEMM  X[8192,128] x W^T[128,4096]  via V_WMMA_F32_16X16X4_F32.
// Block = 256 threads (8 waves); block computes a 16(t) x 1024(col) slab,
// each wave owns 8 contiguous 16x16 output tiles.
// ---------------------------------------------------------------------------
__global__ void __launch_bounds__(256)
gemm_logits_kernel(const float* __restrict__ X,
                   const float* __restrict__ W,
                   const float* __restrict__ b,
                   float* __restrict__ logits) {
  __shared__ float xs[16 * 132];   // 16 t-rows x 128 (padded to kill bank conflicts)
  const int tid  = threadIdx.x;
  const int t0   = blockIdx.x * 16;
  const int cg   = blockIdx.y * 1024;
  const int wave = tid >> 5;
  const int lane = tid & 31;
  const int mrow = lane & 15;      // M for A, N for B/C/D
  const int half = lane >> 4;      // K-half selector
  const int koff = half * 2;

  for (int idx = tid; idx < 16 * 128; idx += 256) {
    int m = idx >> 7, k = idx & 127;
    xs[m * 132 + k] = X[(size_t)(t0 + m) * DX + k];
  }
  __syncthreads();

  const int wcol0 = cg + wave * 128;
  v8f acc[8] = {};

  for (int k0 = 0; k0 < DX; k0 += 4) {
    v2f a;
    a.x = xs[mrow * 132 + k0 + koff];
    a.y = xs[mrow * 132 + k0 + koff + 1];
#pragma unroll
    for (int tile = 0; tile < 8; ++tile) {
      const int col = wcol0 + tile * 16 + mrow;
      v2f bb = *reinterpret_cast<const v2f*>(W + (size_t)col * DX + k0 + koff);
      acc[tile] = __builtin_amdgcn_wmma_f32_16x16x4_f32(
          false, a, false, bb, (short)0, acc[tile], false, false);
    }
  }

#pragma unroll
  for (int tile = 0; tile < 8; ++tile) {
    const int col = wcol0 + tile * 16 + mrow;
    const float bias = b[col];
#pragma unroll
    for (int r = 0; r < 8; ++r) {
      const int m = r + half * 8;                       // t-row within tile
      logits[(size_t)(t0 + m) * KK + col] = acc[tile][r] + bias;
    }
  }
}

// ---------------------------------------------------------------------------
// Kernel 2: row logsumexp  lse[t,i] = logsumexp_j logits[t,i,j]
// One block per t; 4 lanes cooperate per (t,i) row of 64.
// ---------------------------------------------------------------------------
__global__ void __launch_bounds__(256)
row_lse_kernel(const float* __restrict__ logits, float* __restrict__ lse) {
  const int t = blockIdx.x;
  const int tid = threadIdx.x;
  const int r = tid >> 2;          // row i (0..63)
  const int q = tid & 3;           // quarter of row
  const float* p = logits + (size_t)t * KK + r * 64 + q * 16;
  float mx = -INFINITY;
#pragma unroll
  for (int u = 0; u < 16; ++u) mx = fmaxf(mx, p[u]);
  mx = fmaxf(mx, __shfl_xor(mx, 1));
  mx = fmaxf(mx, __shfl_xor(mx, 2));
  float s = 0.f;
#pragma unroll
  for (int u = 0; u < 16; ++u) s += __expf(p[u] - mx);
  s += __shfl_xor(s, 1);
  s += __shfl_xor(s, 2);
  if (q == 0) lse[t * KST + r] = mx + __logf(s);
}

// ---------------------------------------------------------------------------
// Kernel 3: sequential forward recursion (single block, 1024 threads).
// alpha_new[j] = em[t,j] + m + log( sum_i exp(alpha[i]-lse[t,i]-m) * exp(logits[t,i,j]) )
// ---------------------------------------------------------------------------
__global__ void __launch_bounds__(1024)
scan_kernel(const float* __restrict__ init_logits,
            const float* __restrict__ log_em,
            const float* __restrict__ lse,
            const float* __restrict__ logits,
            float* __restrict__ out) {
  __shared__ float alpha[64];
  __shared__ float aprime[64];
  __shared__ float pvec[64];
  __shared__ float sj[64];
  __shared__ float msh;
  const int tid = threadIdx.x;
  const int lane = tid & 31;

  if (tid == 0) {                         // log_softmax normalizer of init_logits
    float mx = -INFINITY;
    for (int k = 0; k < KST; ++k) mx = fmaxf(mx, init_logits[k]);
    float s = 0.f;
    for (int k = 0; k < KST; ++k) s += __expf(init_logits[k] - mx);
    msh = mx + __logf(s);
  }
  __syncthreads();
  if (tid < 64) alpha[tid] = init_logits[tid] - msh + log_em[tid];
  __syncthreads();

  for (int t = 1; t < T_LEN; ++t) {
    const float* lg = logits + (size_t)t * KK;
    if (t + 1 < T_LEN && tid < 128)       // global_prefetch_b8: next step's 16KB
      __builtin_prefetch(logits + (size_t)(t + 1) * KK + tid * 32, 0, 0);

    if (tid < 64) aprime[tid] = alpha[tid] - lse[t * KST + tid];
    __syncthreads();
    if (tid < 32) {                       // m = max_i alpha'[i]
      float v = fmaxf(aprime[lane], aprime[lane + 32]);
      v = fmaxf(v, __shfl_xor(v, 1));
      v = fmaxf(v, __shfl_xor(v, 2));
      v = fmaxf(v, __shfl_xor(v, 4));
      v = fmaxf(v, __shfl_xor(v, 8));
      v = fmaxf(v, __shfl_xor(v, 16));
      if (lane == 0) msh = v;
    }
    __syncthreads();
    const float m = msh;
    if (tid < 64) pvec[tid] = __expf(aprime[tid] - m);
    __syncthreads();

    const int j  = tid >> 4;              // 16 lanes cooperate per output state j
    const int ig = tid & 15;
    float part = 0.f;
#pragma unroll
    for (int u = 0; u < 4; ++u) {
      const int i = ig * 4 + u;
      part += pvec[i] * __expf(lg[i * KST + j]);
    }
    part += __shfl_xor(part, 1);
    part += __shfl_xor(part, 2);
    part += __shfl_xor(part, 4);
    part += __shfl_xor(part, 8);
    if (ig == 0) sj[j] = part;
    __syncthreads();
    if (tid < 64) alpha[tid] = log_em[t * KST + tid] + m + __logf(sj[tid]);
    __syncthreads();
  }

  if (tid == 0) {                         // final logsumexp over alpha
    float mx = -INFINITY;
    for (int k = 0; k < KST; ++k) mx = fmaxf(mx, alpha[k]);
    float s = 0.f;
    for (int k = 0; k < KST; ++k) s += __expf(alpha[k] - mx);
    out[0] = mx + __logf(s);
  }
}

// ---------------------------------------------------------------------------
extern "C" void kernel_launch(void* const* d_in, const int* in_sizes, int n_in,
                              void* d_out, int out_size, void* d_ws, size_t ws_size,
                              hipStream_t stream) {
  const float* X   = (const float*)d_in[0];   // [8192,128]
  const float* Y   = (const float*)d_in[1];   // [8192,64]
  const float* il  = (const float*)d_in[2];   // [64]
  const float* W   = (const float*)d_in[3];   // [64,64,128]
  const float* b   = (const float*)d_in[4];   // [64,64]
  const float* mu  = (const float*)d_in[5];   // [64,64]
  const float* lvu = (const float*)d_in[6];   // [64,64]
  float* out = (float*)d_out;

  float* wsf    = (float*)d_ws;
  float* log_em = wsf;                               // T*K  = 524288 floats
  float* lse    = wsf + (size_t)T_LEN * KST;         // T*K
  float* logits = wsf + 2 * (size_t)T_LEN * KST;     // T*K*K = 33.5M floats

  log_em_kernel<<<T_LEN / 4, 256, 0, stream>>>(Y, mu, lvu, log_em);
  gemm_logits_kernel<<<dim3(T_LEN / 16, 4), 256, 0, stream>>>(X, W, b, logits);
  row_lse_kernel<<<T_LEN, 256, 0, stream>>>(logits, lse);
  scan_kernel<<<1, 1024, 0, stream>>>(il, log_em, lse, logits, out);
}